// AverageContext_15272903705148
// MI455X (gfx1250) — compile-verified
//
#include <hip/hip_runtime.h>

// out[b,t,c] = (1/(t+1)) * sum_{s<=t} x[b,s,c]   (causal running mean)
// x: [16, 4096, 128] f32.  Memory-bound (64 MB HBM traffic ~ 2.7 us at
// 23.3 TB/s): blocked scan.  The 16-row intra-block prefix is computed on the
// matrix engine as L16 x Xtile via V_WMMA_F32_16X16X4_F32 (full f32
// precision), with the cross-block carry injected through the C accumulator.

typedef __attribute__((ext_vector_type(2))) float v2f;
typedef __attribute__((ext_vector_type(8))) float v8f;

#define BATCH 16
#define TLEN  4096
#define CDIM  128
#define NBLK  (TLEN / 16)   // 256 blocks of 16 rows

// ---------------------------------------------------------------------------
// Pass 1: per-block column sums.  bsum[b, blk, c] = sum_{i<16} x[b, blk*16+i, c]
// One thread per (b, blk, c); consecutive c -> fully coalesced.
// ---------------------------------------------------------------------------
__global__ void block_sums(const float* __restrict__ x, float* __restrict__ bsum) {
    int gid = blockIdx.x * blockDim.x + threadIdx.x;          // [0, B*NBLK*C)
    int c   = gid & (CDIM - 1);
    int blk = (gid >> 7) & (NBLK - 1);
    int b   = gid >> (7 + 8);
    const float* p = x + ((size_t)b * TLEN + (size_t)blk * 16) * CDIM + c;
    float s = 0.0f;
#pragma unroll
    for (int i = 0; i < 16; ++i) s += p[(size_t)i * CDIM];
    bsum[gid] = s;
}

// ---------------------------------------------------------------------------
// Pass 2: exclusive scan of the 256 block sums per (b, c).
// 2048 threads total; loads are independent (addresses static), only the
// float add chain is serial.  Data (2 MB) lives in L2.
// ---------------------------------------------------------------------------
__global__ void scan_blocks(const float* __restrict__ bsum, float* __restrict__ excl) {
    int gid = blockIdx.x * blockDim.x + threadIdx.x;          // [0, B*C)
    int c = gid & (CDIM - 1);
    int b = gid >> 7;
    size_t base = (size_t)b * NBLK * CDIM + c;
    float run = 0.0f;
    for (int j = 0; j < NBLK; ++j) {
        excl[base + (size_t)j * CDIM] = run;
        run += bsum[base + (size_t)j * CDIM];
    }
}

// ---------------------------------------------------------------------------
// Pass 3: one wave per 16x16 tile (b, blk, 16-col chunk).
//   D = L16 x Xtile + Carry   via 4 chained V_WMMA_F32_16X16X4_F32,
// then scale row m by 1/(t0+m+1) and store.
//
// Operand layouts (CDNA5 ISA 7.12.2, wave32):
//   A (16x4 f32): lane&15 = M; VGPR0 holds K={0,2}, VGPR1 K={1,3}
//                 (lanes 16-31 take the +2 K half).  L16[m][k] = (k<=m).
//   B (4x16 f32): VGPR0 lanes0-15 = row K=0, lanes16-31 = row K=2;
//                 VGPR1 = rows K={1,3}; cols = lane&15.
//   C/D (16x16 f32): VGPR r: lanes0-15 -> M=r, lanes16-31 -> M=r+8; N=lane&15.
// EXEC is all-ones: grid is an exact fit, no divergence in this kernel.
//
// Scale: divisors t+1 are exact integers <= 4096, so one v_cvt + per-row
// v_add with an exact float literal + v_rcp_f32 (1 ulp) + v_mul replaces the
// full IEEE divide sequence.
// ---------------------------------------------------------------------------
__global__ void tile_prefix_wmma(const float* __restrict__ x,
                                 const float* __restrict__ excl,
                                 float* __restrict__ out) {
    int lane = threadIdx.x & 31;
    int wave = threadIdx.x >> 5;
    int idx  = blockIdx.x * (blockDim.x >> 5) + wave;         // [0, B*NBLK*8)

    int cchunk = idx & 7;                  // 8 chunks of 16 columns
    int blk    = (idx >> 3) & (NBLK - 1);
    int b      = idx >> (3 + 8);

    int t0 = blk * 16;
    int c0 = cchunk * 16;
    int n  = lane & 15;                    // column within tile (and A's M)
    int hi = lane >> 4;                    // 0: lanes 0-15, 1: lanes 16-31

    // Carry from all previous blocks, broadcast down the rows of C.
    float off = excl[((size_t)b * NBLK + blk) * CDIM + c0 + n];
    v8f acc;
#pragma unroll
    for (int r = 0; r < 8; ++r) acc[r] = off;

    const float* xb = x + ((size_t)b * TLEN + t0) * CDIM + c0;

#pragma unroll
    for (int j = 0; j < 4; ++j) {
        int k0 = 4 * j + 2 * hi;           // global K of this operand's VGPR0 slot
        v2f a;                             // L16 chunk, synthesized in registers
        a.x = (k0     <= n) ? 1.0f : 0.0f;
        a.y = (k0 + 1 <= n) ? 1.0f : 0.0f;
        v2f bm;                            // X rows k0, k0+1 at column n
        bm.x = xb[(size_t)(k0    ) * CDIM + n];
        bm.y = xb[(size_t)(k0 + 1) * CDIM + n];
        acc = __builtin_amdgcn_wmma_f32_16x16x4_f32(
            /*neg_a=*/false, a, /*neg_b=*/false, bm,
            /*c_mod=*/(short)0, acc, /*reuse_a=*/false, /*reuse_b=*/false);
    }

    // Scale each row by 1/(t+1) and store per the C/D layout.
    float* ob = out + ((size_t)b * TLEN + t0) * CDIM + c0;
    float dbase = (float)(t0 + 1 + 8 * hi);        // exact in f32 (<= 4089)
#pragma unroll
    for (int r = 0; r < 8; ++r) {
        int   mr  = r + 8 * hi;
        float inv = __builtin_amdgcn_rcpf(dbase + (float)r);   // 1/(t0+mr+1)
        ob[(size_t)mr * CDIM + n] = acc[r] * inv;
    }
}

// ---------------------------------------------------------------------------
extern "C" void kernel_launch(void* const* d_in, const int* in_sizes, int n_in,
                              void* d_out, int out_size, void* d_ws, size_t ws_size,
                              hipStream_t stream) {
    const float* x  = (const float*)d_in[0];
    float* out      = (float*)d_out;
    float* bsum     = (float*)d_ws;                                   // 2 MB
    float* excl     = bsum + (size_t)BATCH * NBLK * CDIM;             // 2 MB

    // Pass 1: B*NBLK*C = 524288 threads
    block_sums<<<(BATCH * NBLK * CDIM) / 256, 256, 0, stream>>>(x, bsum);
    // Pass 2: B*C = 2048 threads
    scan_blocks<<<(BATCH * CDIM) / 256, 256, 0, stream>>>(bsum, excl);
    // Pass 3: B*NBLK*8 = 32768 waves, 8 waves per block
    tile_prefix_wmma<<<(BATCH * NBLK * 8) / 8, 256, 0, stream>>>(x, excl, out);
}